// GraphNetwork_GATv2_EdgeFeat_962072674439
// MI455X (gfx1250) — compile-verified
//
#include <hip/hip_runtime.h>
#include <math.h>

#define NN    16384
#define FDIM  128
#define RDIM  16
#define HEADS 8
#define CDIM  64
#define HC    512
#define NE    163840
#define EPT   (NE + NN)          // 180224 edges incl. self loops
#define NEG_SLOPE 0.2f
#define SELF_FILL 0.5f

typedef float v2f __attribute__((ext_vector_type(2)));
typedef float v8f __attribute__((ext_vector_type(8)));

// ---------------------------------------------------------------------------
// fp32 WMMA GEMM:  Y[M,Nc] = X[M,K] @ W[K,Nc] + b[Nc]
// 256 threads = 8 waves; wave w computes a 16(M) x 64(N) strip:
// one A fragment reused across 4 B fragments / 4 accumulators.
// Grid: (Nc/64, M/128). Requires M%128==0, Nc%64==0, K%4==0.
// ---------------------------------------------------------------------------
__global__ void __launch_bounds__(256)
wmma_gemm_bias(const float* __restrict__ X, const float* __restrict__ W,
               const float* __restrict__ b, float* __restrict__ Y,
               int M, int K, int Nc)
{
    const int wave = threadIdx.x >> 5;
    const int lane = threadIdx.x & 31;
    const int mt   = (blockIdx.y * 8 + wave) * 16;
    const int nt   = blockIdx.x * 64;
    const int r    = lane & 15;
    const int hsel = lane >> 4;           // 0: K pair {0,1}, 1: K pair {2,3}

    const float* Xrow = X + (size_t)(mt + r) * K;
    const int    col0 = nt + r;

    v8f c0 = {}, c1 = {}, c2 = {}, c3 = {};
    for (int k0 = 0; k0 < K; k0 += 4) {
        v2f a;
        a.x = Xrow[k0 + 2 * hsel];
        a.y = Xrow[k0 + 2 * hsel + 1];
        const float* Wr0 = W + (size_t)(k0 + 2 * hsel)     * Nc;
        const float* Wr1 = W + (size_t)(k0 + 2 * hsel + 1) * Nc;
        v2f b0, b1, b2, b3;
        b0.x = Wr0[col0];      b0.y = Wr1[col0];
        b1.x = Wr0[col0 + 16]; b1.y = Wr1[col0 + 16];
        b2.x = Wr0[col0 + 32]; b2.y = Wr1[col0 + 32];
        b3.x = Wr0[col0 + 48]; b3.y = Wr1[col0 + 48];
        c0 = __builtin_amdgcn_wmma_f32_16x16x4_f32(false, a, false, b0, (short)0, c0, false, false);
        c1 = __builtin_amdgcn_wmma_f32_16x16x4_f32(false, a, false, b1, (short)0, c1, false, false);
        c2 = __builtin_amdgcn_wmma_f32_16x16x4_f32(false, a, false, b2, (short)0, c2, false, false);
        c3 = __builtin_amdgcn_wmma_f32_16x16x4_f32(false, a, false, b3, (short)0, c3, false, false);
    }
#pragma unroll
    for (int t = 0; t < 4; ++t) {
        const v8f   c    = (t == 0) ? c0 : (t == 1) ? c1 : (t == 2) ? c2 : c3;
        const int   col  = col0 + 16 * t;
        const float bias = b[col];
#pragma unroll
        for (int v = 0; v < 8; ++v) {
            const int row = mt + v + 8 * hsel;
            Y[(size_t)row * Nc + col] = c[v] + bias;
        }
    }
}

// ---------------------------------------------------------------------------
// init helpers
// ---------------------------------------------------------------------------
__global__ void init_rows_bias(float* __restrict__ acc, const float* __restrict__ bias)
{
    unsigned gid = blockIdx.x * blockDim.x + threadIdx.x;
    if (gid < (unsigned)NN * HC) acc[gid] = bias[gid & (HC - 1)];
}

__global__ void fill_value(float* __restrict__ p, float v, unsigned n)
{
    unsigned gid = blockIdx.x * blockDim.x + threadIdx.x;
    if (gid < n) p[gid] = v;
}

// ---------------------------------------------------------------------------
// float atomic max via int/uint trick (maxv initialized to -inf)
// ---------------------------------------------------------------------------
__device__ __forceinline__ void atomicMaxFloat(float* addr, float val)
{
    if (val >= 0.0f) atomicMax((int*)addr, __float_as_int(val));
    else             atomicMin((unsigned int*)addr, __float_as_uint(val));
}

// ---------------------------------------------------------------------------
// logits: one wave per (edge, head); ep = e@We computed on the fly (K=16)
// ---------------------------------------------------------------------------
__global__ void __launch_bounds__(256)
edge_logits(const int* __restrict__ ei, const float* __restrict__ eattr,
            const float* __restrict__ xl, const float* __restrict__ xr,
            const float* __restrict__ We, const float* __restrict__ att,
            float* __restrict__ logitsb, float* __restrict__ maxv)
{
    const unsigned wid  = blockIdx.x * 8 + (threadIdx.x >> 5);
    const unsigned lane = threadIdx.x & 31;
    if (wid >= (unsigned)EPT * HEADS) return;
    const unsigned e = wid >> 3, h = wid & 7;

    int src, dst;
    float ea[RDIM];
    if (e < (unsigned)NE) {
        src = ei[e];
        dst = ei[NE + e];
#pragma unroll
        for (int q = 0; q < RDIM; ++q) ea[q] = eattr[e * RDIM + q];
    } else {
        src = dst = (int)(e - NE);
#pragma unroll
        for (int q = 0; q < RDIM; ++q) ea[q] = SELF_FILL;
    }

    const float* xls = xl + (size_t)src * HC + h * CDIM;
    const float* xrd = xr + (size_t)dst * HC + h * CDIM;

    float part = 0.0f;
#pragma unroll
    for (int i = 0; i < 2; ++i) {
        const int c = (int)lane + 32 * i;
        float epv = 0.0f;
#pragma unroll
        for (int q = 0; q < RDIM; ++q)
            epv += ea[q] * We[q * HC + h * CDIM + c];
        float zv = xls[c] + xrd[c] + epv;
        zv = zv > 0.0f ? zv : NEG_SLOPE * zv;
        part += zv * att[h * CDIM + c];
    }
#pragma unroll
    for (int off = 16; off > 0; off >>= 1)
        part += __shfl_down(part, off, 32);

    if (lane == 0) {
        logitsb[e * HEADS + h] = part;
        atomicMaxFloat(&maxv[dst * HEADS + h], part);
    }
}

// ---------------------------------------------------------------------------
// exp(logit - max) and denominator accumulation; one thread per (edge, head)
// ---------------------------------------------------------------------------
__global__ void __launch_bounds__(256)
edge_exp(const int* __restrict__ ei, float* __restrict__ logitsb,
         const float* __restrict__ maxv, float* __restrict__ denom)
{
    unsigned gid = blockIdx.x * blockDim.x + threadIdx.x;
    if (gid >= (unsigned)EPT * HEADS) return;
    const unsigned e = gid >> 3, h = gid & 7;
    const int dst = (e < (unsigned)NE) ? ei[NE + e] : (int)(e - NE);
    const float ex = expf(logitsb[gid] - maxv[dst * HEADS + h]);
    logitsb[gid] = ex;
    atomicAdd(&denom[dst * HEADS + h], ex);
}

// ---------------------------------------------------------------------------
// aggregation: one thread per (edge, head, channel) -> coalesced scatter
// ---------------------------------------------------------------------------
__global__ void __launch_bounds__(256)
edge_aggregate(const int* __restrict__ ei, const float* __restrict__ exbuf,
               const float* __restrict__ denom, const float* __restrict__ xl,
               float* __restrict__ acc)
{
    unsigned gid = blockIdx.x * blockDim.x + threadIdx.x;
    if (gid >= (unsigned)EPT * HC) return;
    const unsigned c = gid & (CDIM - 1);
    const unsigned h = (gid >> 6) & (HEADS - 1);
    const unsigned e = gid >> 9;
    int src, dst;
    if (e < (unsigned)NE) { src = ei[e]; dst = ei[NE + e]; }
    else                  { src = dst = (int)(e - NE); }
    const float alpha = exbuf[e * HEADS + h] / (denom[dst * HEADS + h] + 1e-16f);
    atomicAdd(&acc[(size_t)dst * HC + h * CDIM + c],
              xl[(size_t)src * HC + h * CDIM + c] * alpha);
}

// ---------------------------------------------------------------------------
// layer-2 finalize: mean over heads + bias2
// ---------------------------------------------------------------------------
__global__ void __launch_bounds__(256)
finalize_mean(const float* __restrict__ acc2, const float* __restrict__ bias2,
              float* __restrict__ out)
{
    unsigned gid = blockIdx.x * blockDim.x + threadIdx.x;
    if (gid >= (unsigned)NN * CDIM) return;
    const unsigned n = gid >> 6, c = gid & (CDIM - 1);
    float s = 0.0f;
#pragma unroll
    for (int h = 0; h < HEADS; ++h)
        s += acc2[(size_t)n * HC + h * CDIM + c];
    out[gid] = s * (1.0f / HEADS) + bias2[c];
}

// ---------------------------------------------------------------------------
extern "C" void kernel_launch(void* const* d_in, const int* in_sizes, int n_in,
                              void* d_out, int out_size, void* d_ws, size_t ws_size,
                              hipStream_t stream)
{
    const float* x     = (const float*)d_in[0];
    const int*   ei    = (const int*)  d_in[1];
    const float* eattr = (const float*)d_in[2];
    const float* Wl1   = (const float*)d_in[3];
    const float* bl1   = (const float*)d_in[4];
    const float* Wr1   = (const float*)d_in[5];
    const float* br1   = (const float*)d_in[6];
    const float* We1   = (const float*)d_in[7];
    const float* att1  = (const float*)d_in[8];
    const float* bias1 = (const float*)d_in[9];
    const float* Wl2   = (const float*)d_in[10];
    const float* bl2   = (const float*)d_in[11];
    const float* Wr2   = (const float*)d_in[12];
    const float* br2   = (const float*)d_in[13];
    const float* We2   = (const float*)d_in[14];
    const float* att2  = (const float*)d_in[15];
    const float* bias2 = (const float*)d_in[16];
    float* out = (float*)d_out;

    const size_t NODE = (size_t)NN * HC;           // 8,388,608 floats
    float* ws      = (float*)d_ws;
    float* xl1     = ws;
    float* xr1     = xl1 + NODE;
    float* acc1    = xr1 + NODE;                   // layer-1 output (h1)
    float* xl2     = acc1 + NODE;
    float* xr2     = xl2 + NODE;
    float* acc2    = xr2 + NODE;
    float* logitsb = acc2 + NODE;                  // EPT*HEADS
    float* maxv    = logitsb + (size_t)EPT * HEADS;
    float* denom   = maxv + (size_t)NN * HEADS;

    const unsigned EH   = (unsigned)EPT * HEADS;   // 1,441,792
    const unsigned EHC  = (unsigned)EPT * HC;      // 92,274,688
    const unsigned NH   = (unsigned)NN * HEADS;    // 131,072
    const dim3 blk(256);
    const dim3 gemmGrid(HC / 64, NN / 128);        // (8, 128)

    // ---------------- layer 1 ----------------
    init_rows_bias<<<(NN * HC + 255) / 256, blk, 0, stream>>>(acc1, bias1);
    fill_value<<<(NH + 255) / 256, blk, 0, stream>>>(maxv, -INFINITY, NH);
    fill_value<<<(NH + 255) / 256, blk, 0, stream>>>(denom, 0.0f, NH);
    fill_value<<<(NN * HC + 255) / 256, blk, 0, stream>>>(acc2, 0.0f, NN * HC);

    wmma_gemm_bias<<<gemmGrid, blk, 0, stream>>>(x, Wl1, bl1, xl1, NN, FDIM, HC);
    wmma_gemm_bias<<<gemmGrid, blk, 0, stream>>>(x, Wr1, br1, xr1, NN, FDIM, HC);

    edge_logits<<<EH / 8, blk, 0, stream>>>(ei, eattr, xl1, xr1, We1, att1, logitsb, maxv);
    edge_exp<<<(EH + 255) / 256, blk, 0, stream>>>(ei, logitsb, maxv, denom);
    edge_aggregate<<<(EHC + 255) / 256, blk, 0, stream>>>(ei, logitsb, denom, xl1, acc1);

    // ---------------- layer 2 ----------------
    fill_value<<<(NH + 255) / 256, blk, 0, stream>>>(maxv, -INFINITY, NH);
    fill_value<<<(NH + 255) / 256, blk, 0, stream>>>(denom, 0.0f, NH);

    wmma_gemm_bias<<<gemmGrid, blk, 0, stream>>>(acc1, Wl2, bl2, xl2, NN, HC, HC);
    wmma_gemm_bias<<<gemmGrid, blk, 0, stream>>>(acc1, Wr2, br2, xr2, NN, HC, HC);

    edge_logits<<<EH / 8, blk, 0, stream>>>(ei, eattr, xl2, xr2, We2, att2, logitsb, maxv);
    edge_exp<<<(EH + 255) / 256, blk, 0, stream>>>(ei, logitsb, maxv, denom);
    edge_aggregate<<<(EHC + 255) / 256, blk, 0, stream>>>(ei, logitsb, denom, xl2, acc2);

    finalize_mean<<<(NN * CDIM + 255) / 256, blk, 0, stream>>>(acc2, bias2, out);
}